// SEGNN_28991029248405
// MI455X (gfx1250) — compile-verified
//
#include <hip/hip_runtime.h>

// ---------------------------------------------------------------------------
// SEGNN on MI455X (gfx1250). All tensor-product weight GEMMs run on
// V_WMMA_F32_16X16X4_F32 (full fp32 matrix pipe, wave32). One wave owns a
// 16-item tile; per-wave-private LDS tiles, so no barriers are required
// (LDS ops are in-order within a wave). Weights are pre-packed into the WMMA
// B-fragment pair layout so each B fragment is a single global b64 load.
// ---------------------------------------------------------------------------

typedef __attribute__((ext_vector_type(2))) float v2f;
typedef __attribute__((ext_vector_type(4))) float v4f;
typedef __attribute__((ext_vector_type(8))) float v8f;

#define DEVI __device__ __forceinline__

DEVI v8f wmma_f32_16x16x4(v2f a, v2f b, v8f c) {
  // 8 args: (neg_a, A, neg_b, B, c_mod, C, reuse_a, reuse_b)
  return __builtin_amdgcn_wmma_f32_16x16x4_f32(false, a, false, b, (short)0, c,
                                               false, false);
}

DEVI float sig_(float x) { return 1.0f / (1.0f + __expf(-x)); }

struct TPW {
  const v2f* wss;  // packed [C0I/2][C0O] pairs
  const v2f* wvs;  // packed [C1I/2][C0O]
  const v2f* wsv;  // packed [C0I/2][C1O]
  const v2f* wv0;  // packed [C1I/2][C1O]
  const v2f* wv1;  // packed [C1I/2][C1O]
  const float* b;  // [C0O]
};

// ---------------------------------------------------------------------------
// Generic 16-item O3 tensor product (+optional e3nn gate), WMMA-tiled.
// Sin: LDS [16][C0I]; Vin: LDS planes [3][16][C1I]; At: LDS [16][3].
// Outputs: Sout [16][SC] (SC = gated ? C1O : C0O), Vout planes [3][16][C1O].
// ---------------------------------------------------------------------------
template <int C0I, int C1I, int C0O, int C1O, bool GATE>
DEVI void tp16(const float* __restrict__ Sin, const float* __restrict__ Vin,
               float* __restrict__ Sout, float* __restrict__ Vout,
               const float* __restrict__ At, TPW w) {
  constexpr int NB0 = C0O / 16;  // scalar output blocks
  constexpr int NB1 = C1O / 16;  // vector output blocks
  constexpr int KS0 = C0I / 4;   // k-steps over scalar input
  constexpr int KS1 = C1I / 4;   // k-steps over vector input
  constexpr int SC = GATE ? C1O : C0O;
  constexpr int NBS = SC / 16;

  const int lane = threadIdx.x & 31;
  const int e = lane & 15;        // A-fragment row (item)
  const int h = lane >> 4;        // k-half select
  const int n = lane & 15;        // column within output block
  const int rowoff = h * 8;       // C/D rows: lanes>=16 hold M=r+8

  const float aE0 = At[e * 3 + 0], aE1 = At[e * 3 + 1], aE2 = At[e * 3 + 2];
  float aM[8][3];
#pragma unroll
  for (int r = 0; r < 8; ++r)
#pragma unroll
    for (int d = 0; d < 3; ++d) aM[r][d] = At[(r + rowoff) * 3 + d];

  // dots = v . a, computed directly in WMMA A-fragment layout (registers).
  v2f df[KS1];
#pragma unroll
  for (int kk = 0; kk < KS1; ++kk) {
    const int c0 = 4 * kk + 2 * h;
    const float* p0 = &Vin[(0 * 16 + e) * C1I + c0];
    const float* p1 = &Vin[(1 * 16 + e) * C1I + c0];
    const float* p2 = &Vin[(2 * 16 + e) * C1I + c0];
    v2f t;
    t[0] = p0[0] * aE0 + p1[0] * aE1 + p2[0] * aE2;
    t[1] = p0[1] * aE0 + p1[1] * aE1 + p2[1] * aE2;
    df[kk] = t;
  }

  // ---- scalar path: GS = Sin*wss + dots*wvs + b ----
  v8f gs[NB0];
#pragma unroll
  for (int ob = 0; ob < NB0; ++ob) {
    const float bb = w.b[ob * 16 + n];
#pragma unroll
    for (int r = 0; r < 8; ++r) gs[ob][r] = bb;
  }
#pragma unroll
  for (int kk = 0; kk < KS0; ++kk) {
    const v2f af = *(const v2f*)&Sin[e * C0I + 4 * kk + 2 * h];
#pragma unroll
    for (int ob = 0; ob < NB0; ++ob)
      gs[ob] = wmma_f32_16x16x4(af, w.wss[(2 * kk + h) * C0O + ob * 16 + n], gs[ob]);
  }
#pragma unroll
  for (int kk = 0; kk < KS1; ++kk) {
#pragma unroll
    for (int ob = 0; ob < NB0; ++ob)
      gs[ob] = wmma_f32_16x16x4(df[kk], w.wvs[(2 * kk + h) * C0O + ob * 16 + n], gs[ob]);
  }
#pragma unroll
  for (int ob = 0; ob < NBS; ++ob)
#pragma unroll
    for (int r = 0; r < 8; ++r) {
      const float x = gs[ob][r];
      Sout[(r + rowoff) * SC + ob * 16 + n] = GATE ? (x * sig_(x)) : x;
    }

  // ---- vector path. Cross term folded into P_d = v_d * wv1 GEMMs:
  //      out_d = U_d + T*a_d + a_{d+2}*P_{d+1} - a_{d+1}*P_{d+2}.
#pragma unroll
  for (int ob = 0; ob < NB1; ++ob) {
    v8f T;
#pragma unroll
    for (int r = 0; r < 8; ++r) T[r] = 0.0f;
#pragma unroll
    for (int kk = 0; kk < KS0; ++kk) {
      const v2f af = *(const v2f*)&Sin[e * C0I + 4 * kk + 2 * h];
      T = wmma_f32_16x16x4(af, w.wsv[(2 * kk + h) * C1O + ob * 16 + n], T);
    }
    v8f U[3], P[3];
#pragma unroll
    for (int d = 0; d < 3; ++d)
#pragma unroll
      for (int r = 0; r < 8; ++r) {
        U[d][r] = 0.0f;
        P[d][r] = 0.0f;
      }
#pragma unroll
    for (int d = 0; d < 3; ++d)
#pragma unroll
      for (int kk = 0; kk < KS1; ++kk) {
        const v2f af = *(const v2f*)&Vin[(d * 16 + e) * C1I + 4 * kk + 2 * h];
        U[d] = wmma_f32_16x16x4(af, w.wv0[(2 * kk + h) * C1O + ob * 16 + n], U[d]);
        P[d] = wmma_f32_16x16x4(af, w.wv1[(2 * kk + h) * C1O + ob * 16 + n], P[d]);
      }
#pragma unroll
    for (int r = 0; r < 8; ++r) {
      float g = 1.0f;
      if constexpr (GATE) g = sig_(gs[NB1 + ob][r]);  // gate scalars = cols [H, H+C1O)
#pragma unroll
      for (int d = 0; d < 3; ++d) {
        const int d1 = (d + 1) % 3, d2 = (d + 2) % 3;
        const float val = U[d][r] + T[r] * aM[r][d] + aM[r][d2] * P[d1][r] -
                          aM[r][d1] * P[d2][r];
        Vout[(d * 16 + r + rowoff) * C1O + ob * 16 + n] = g * val;
      }
    }
  }
}

// ---------------------------------------------------------------------------
// Per-wave LDS tile buffers (no barriers needed: one wave per tile).
// ---------------------------------------------------------------------------
struct alignas(16) TileBuf64 {
  float S1[16 * 64];
  float V1[3 * 16 * 64];
  float S2[16 * 32];
  float V2[3 * 16 * 32];
  float At[64];
};
struct alignas(16) TileBuf32 {
  float S1[16 * 32];
  float V1[3 * 16 * 32];
  float S2[16 * 32];
  float V2[3 * 16 * 32];
  float At[64];
};

#define WPB 2  // waves per block; 2 x 24.8KB static LDS = 49.7KB

// stage one node row (32 scalars + 32x3 vectors) into one half of the 64-wide
// tiles; lane pair (eu, half) handles (item eu, sender/receiver half).
DEVI void stage_half64(const float* __restrict__ srow, const float* __restrict__ vrow,
                       float* __restrict__ S1, float* __restrict__ V1, int eu, int half) {
  const v4f* sp = (const v4f*)srow;
#pragma unroll
  for (int j = 0; j < 8; ++j) *(v4f*)&S1[eu * 64 + half * 32 + j * 4] = sp[j];
  const v4f* vp = (const v4f*)vrow;
#pragma unroll
  for (int f = 0; f < 24; ++f) {
    const v4f q = vp[f];
#pragma unroll
    for (int j = 0; j < 4; ++j) {
      const int g = f * 4 + j;
      const int c = g / 3, d = g - 3 * c;
      V1[(d * 16 + eu) * 64 + half * 32 + c] = q[j];
    }
  }
}

// ---------------------------------------------------------------------------
// Edge message kernel: gather -> msg0 gate -> msg1 gate -> atomic segment-sum.
// ---------------------------------------------------------------------------
__global__ __launch_bounds__(WPB * 32) void edge_msg_kernel(
    const float* __restrict__ sN, const float* __restrict__ vN,
    const float* __restrict__ evec, const int* __restrict__ snd,
    const int* __restrict__ rcv, float* __restrict__ ags, float* __restrict__ agv,
    TPW w0, TPW w1, int nTiles) {
  __shared__ TileBuf64 lds[WPB];
  const int wave = threadIdx.x >> 5;
  const int lane = threadIdx.x & 31;
  const int tile = blockIdx.x * WPB + wave;
  if (tile >= nTiles) return;  // wave-uniform exit; EXEC stays all-1
  TileBuf64& L = lds[wave];
  const int e0 = tile * 16;

  const int eu = lane >> 1, half = lane & 1;
  const int edge = e0 + eu;
  const int rnode = rcv[edge];
  const int node = half ? rnode : snd[edge];
  stage_half64(sN + (size_t)node * 32, vN + (size_t)node * 96, L.S1, L.V1, eu, half);
  if (lane < 16) {
#pragma unroll
    for (int d = 0; d < 3; ++d) L.At[lane * 3 + d] = evec[(size_t)(e0 + lane) * 3 + d];
  }

  tp16<64, 64, 64, 32, true>(L.S1, L.V1, L.S2, L.V2, L.At, w0);   // msg0 gate
  tp16<32, 32, 64, 32, true>(L.S2, L.V2, L.S1, L.V1, L.At, w1);   // msg1 gate

  // segment_sum via f32 atomics (aggregates are L2 resident: ~26MB << 192MB)
  const int cb = half * 16;
#pragma unroll
  for (int c = cb; c < cb + 16; ++c) {
    atomicAdd(&ags[(size_t)rnode * 32 + c], L.S1[eu * 32 + c]);
#pragma unroll
    for (int d = 0; d < 3; ++d)
      atomicAdd(&agv[((size_t)rnode * 32 + c) * 3 + d], L.V1[(d * 16 + eu) * 32 + c]);
  }
}

// ---------------------------------------------------------------------------
// Node update: [s|ags],[v|agv] -> upd_gate -> upd_lin -> residual (in place).
// ---------------------------------------------------------------------------
__global__ __launch_bounds__(WPB * 32) void node_update_kernel(
    float* __restrict__ sN, float* __restrict__ vN, const float* __restrict__ ags,
    const float* __restrict__ agv, const float* __restrict__ nattr, TPW wg, TPW wl,
    int nTiles) {
  __shared__ TileBuf64 lds[WPB];
  const int wave = threadIdx.x >> 5;
  const int lane = threadIdx.x & 31;
  const int tile = blockIdx.x * WPB + wave;
  if (tile >= nTiles) return;
  TileBuf64& L = lds[wave];

  const int eu = lane >> 1, half = lane & 1;
  const int node = tile * 16 + eu;
  const float* sr = half ? (ags + (size_t)node * 32) : (sN + (size_t)node * 32);
  const float* vr = half ? (agv + (size_t)node * 96) : (vN + (size_t)node * 96);
  stage_half64(sr, vr, L.S1, L.V1, eu, half);
  if (lane < 16) {
#pragma unroll
    for (int d = 0; d < 3; ++d)
      L.At[lane * 3 + d] = nattr[(size_t)(tile * 16 + lane) * 3 + d];
  }

  tp16<64, 64, 64, 32, true>(L.S1, L.V1, L.S2, L.V2, L.At, wg);    // upd_gate
  tp16<32, 32, 32, 32, false>(L.S2, L.V2, L.S1, L.V1, L.At, wl);   // upd_lin

  const int cb = half * 16;
#pragma unroll
  for (int c = cb; c < cb + 16; ++c) {
    sN[(size_t)node * 32 + c] += L.S1[eu * 32 + c];
#pragma unroll
    for (int d = 0; d < 3; ++d)
      vN[((size_t)node * 32 + c) * 3 + d] += L.V1[(d * 16 + eu) * 32 + c];
  }
}

// ---------------------------------------------------------------------------
// Decoder: dec_gate (WMMA) + dec_out (C1O=1, VALU) -> out [N][3].
// ---------------------------------------------------------------------------
__global__ __launch_bounds__(WPB * 32) void decode_kernel(
    const float* __restrict__ sN, const float* __restrict__ vN,
    const float* __restrict__ nattr, TPW wg, const float* __restrict__ wsvO,
    const float* __restrict__ wv0O, const float* __restrict__ wv1O,
    float* __restrict__ out, int nTiles) {
  __shared__ TileBuf32 lds[WPB];
  const int wave = threadIdx.x >> 5;
  const int lane = threadIdx.x & 31;
  const int tile = blockIdx.x * WPB + wave;
  if (tile >= nTiles) return;
  TileBuf32& L = lds[wave];

  const int eu = lane >> 1, half = lane & 1;
  const int node = tile * 16 + eu;
  const v4f* sp = (const v4f*)(sN + (size_t)node * 32);
#pragma unroll
  for (int j = 0; j < 4; ++j)
    *(v4f*)&L.S1[eu * 32 + half * 16 + j * 4] = sp[half * 4 + j];
  const v4f* vp = (const v4f*)(vN + (size_t)node * 96);
#pragma unroll
  for (int f = 0; f < 12; ++f) {
    const v4f q = vp[half * 12 + f];
#pragma unroll
    for (int j = 0; j < 4; ++j) {
      const int g = half * 48 + f * 4 + j;
      const int c = g / 3, d = g - 3 * c;
      L.V1[(d * 16 + eu) * 32 + c] = q[j];
    }
  }
  if (lane < 16) {
#pragma unroll
    for (int d = 0; d < 3; ++d)
      L.At[lane * 3 + d] = nattr[(size_t)(tile * 16 + lane) * 3 + d];
  }

  tp16<32, 32, 64, 32, true>(L.S1, L.V1, L.S2, L.V2, L.At, wg);  // dec_gate

  if (lane < 16) {  // dec_out: 1x1o output, trivial VALU reduction
    const int nd = tile * 16 + lane;
    const float a[3] = {L.At[lane * 3 + 0], L.At[lane * 3 + 1], L.At[lane * 3 + 2]};
    float T = 0.f, U[3] = {0.f, 0.f, 0.f}, P[3] = {0.f, 0.f, 0.f};
#pragma unroll
    for (int c = 0; c < 32; ++c) {
      T += L.S2[lane * 32 + c] * wsvO[c];
#pragma unroll
      for (int d = 0; d < 3; ++d) {
        const float vv = L.V2[(d * 16 + lane) * 32 + c];
        U[d] += vv * wv0O[c];
        P[d] += vv * wv1O[c];
      }
    }
#pragma unroll
    for (int d = 0; d < 3; ++d) {
      const int d1 = (d + 1) % 3, d2 = (d + 2) % 3;
      out[(size_t)nd * 3 + d] = U[d] + T * a[d] + a[d2] * P[d1] - a[d1] * P[d2];
    }
  }
}

// ---------------------------------------------------------------------------
// Embedding (8,2 -> 32,32): tiny, one thread per node (0.14 GFLOP total).
// ---------------------------------------------------------------------------
__global__ void embed_kernel(const float* __restrict__ ns, const float* __restrict__ nv,
                             const float* __restrict__ na, const float* __restrict__ b,
                             const float* __restrict__ wss, const float* __restrict__ wsv,
                             const float* __restrict__ wv0, const float* __restrict__ wv1,
                             const float* __restrict__ wvs, float* __restrict__ sOut,
                             float* __restrict__ vOut, int n) {
  const int i = blockIdx.x * blockDim.x + threadIdx.x;
  if (i >= n) return;
  float s[8];
#pragma unroll
  for (int c = 0; c < 8; ++c) s[c] = ns[(size_t)i * 8 + c];
  float v[2][3];
#pragma unroll
  for (int c = 0; c < 2; ++c)
#pragma unroll
    for (int d = 0; d < 3; ++d) v[c][d] = nv[(size_t)i * 6 + c * 3 + d];
  float a[3];
#pragma unroll
  for (int d = 0; d < 3; ++d) a[d] = na[(size_t)i * 3 + d];
  float dot[2], cr[2][3];
#pragma unroll
  for (int c = 0; c < 2; ++c) {
    dot[c] = v[c][0] * a[0] + v[c][1] * a[1] + v[c][2] * a[2];
#pragma unroll
    for (int d = 0; d < 3; ++d) {
      const int d1 = (d + 1) % 3, d2 = (d + 2) % 3;
      cr[c][d] = v[c][d1] * a[d2] - v[c][d2] * a[d1];
    }
  }
#pragma unroll 4
  for (int o = 0; o < 32; ++o) {
    float so = b[o];
#pragma unroll
    for (int c = 0; c < 8; ++c) so += s[c] * wss[c * 32 + o];
    so += dot[0] * wvs[o] + dot[1] * wvs[32 + o];
    sOut[(size_t)i * 32 + o] = so;
    float T = 0.f;
#pragma unroll
    for (int c = 0; c < 8; ++c) T += s[c] * wsv[c * 32 + o];
#pragma unroll
    for (int d = 0; d < 3; ++d) {
      float vo = T * a[d];
#pragma unroll
      for (int c = 0; c < 2; ++c)
        vo += v[c][d] * wv0[c * 32 + o] + cr[c][d] * wv1[c * 32 + o];
      vOut[((size_t)i * 32 + o) * 3 + d] = vo;
    }
  }
}

// Pack weight W[K][N] (row-major) into B-fragment pair layout:
// out[kk*N + n] = {W[2kk][n], W[2kk+1][n]}  (one b64 load per B fragment).
__global__ void pack_pairs_kernel(const float* __restrict__ W, float* __restrict__ out,
                                  int K, int N) {
  const int i = blockIdx.x * blockDim.x + threadIdx.x;
  const int tot = (K / 2) * N;
  if (i >= tot) return;
  const int kk = i / N, n = i - kk * N;
  out[2 * i + 0] = W[(2 * kk + 0) * N + n];
  out[2 * i + 1] = W[(2 * kk + 1) * N + n];
}

// ---------------------------------------------------------------------------
// Host driver. Input order: top-level setup_inputs() insertion order, with
// 'params' flattened jax-tree style (dict keys sorted at every level):
//   0 node_s, 1 node_v, 2 edge_vec, 3 node_attr_v,
//   4.. params: groups {dec_gate, dec_out, embed, L0.msg0, L0.msg1,
//   L0.upd_gate, L0.upd_lin, L1.msg0, L1.msg1, L1.upd_gate, L1.upd_lin},
//   each group = {b, wss, wsv, wv0, wv1, wvs},
//   70 senders, 71 receivers.
// ---------------------------------------------------------------------------
extern "C" void kernel_launch(void* const* d_in, const int* in_sizes, int n_in,
                              void* d_out, int out_size, void* d_ws, size_t ws_size,
                              hipStream_t stream) {
  (void)in_sizes; (void)n_in; (void)out_size; (void)ws_size;
  constexpr int N = 50000, E = 400000;
  const float* node_s = (const float*)d_in[0];
  const float* node_v = (const float*)d_in[1];
  const float* edge_vec = (const float*)d_in[2];
  const float* nattr = (const float*)d_in[3];
  const int* senders = (const int*)d_in[70];
  const int* receivers = (const int*)d_in[71];
  auto P = [&](int g, int j) -> const float* { return (const float*)d_in[4 + g * 6 + j]; };
  // group ids: 0=dec_gate 1=dec_out 2=embed 3..6=L0{msg0,msg1,updg,updl} 7..10=L1

  float* ws = (float*)d_ws;
  size_t off = 0;
  auto take = [&](size_t nf) { size_t o = off; off += (nf + 7) & ~(size_t)7; return o; };
  float* sBuf = ws + take((size_t)N * 32);
  float* vBuf = ws + take((size_t)N * 96);
  float* agsB = ws + take((size_t)N * 32);
  float* agvB = ws + take((size_t)N * 96);  // contiguous with agsB

  auto packM = [&](const float* W, int K, int Nc) -> const v2f* {
    float* o = ws + take((size_t)K * Nc);
    const int tot = (K / 2) * Nc;
    pack_pairs_kernel<<<(tot + 255) / 256, 256, 0, stream>>>(W, o, K, Nc);
    return (const v2f*)o;
  };
  auto packG = [&](int g, int c0i, int c1i, int c0o, int c1o) -> TPW {
    TPW t;
    t.wss = packM(P(g, 1), c0i, c0o);
    t.wvs = packM(P(g, 5), c1i, c0o);
    t.wsv = packM(P(g, 2), c0i, c1o);
    t.wv0 = packM(P(g, 3), c1i, c1o);
    t.wv1 = packM(P(g, 4), c1i, c1o);
    t.b = P(g, 0);
    return t;
  };

  const TPW dec_gate = packG(0, 32, 32, 64, 32);
  const TPW m0[2] = {packG(3, 64, 64, 64, 32), packG(7, 64, 64, 64, 32)};
  const TPW m1[2] = {packG(4, 32, 32, 64, 32), packG(8, 32, 32, 64, 32)};
  const TPW ug[2] = {packG(5, 64, 64, 64, 32), packG(9, 64, 64, 64, 32)};
  const TPW ul[2] = {packG(6, 32, 32, 32, 32), packG(10, 32, 32, 32, 32)};

  embed_kernel<<<(N + 255) / 256, 256, 0, stream>>>(
      node_s, node_v, nattr, P(2, 0), P(2, 1), P(2, 2), P(2, 3), P(2, 4), P(2, 5),
      sBuf, vBuf, N);

  const int eTiles = E / 16, nTiles = N / 16;
  const int eBlocks = (eTiles + WPB - 1) / WPB;
  const int nBlocks = (nTiles + WPB - 1) / WPB;

  for (int l = 0; l < 2; ++l) {
    hipMemsetAsync(agsB, 0, (size_t)N * (32 + 96) * sizeof(float), stream);
    edge_msg_kernel<<<eBlocks, WPB * 32, 0, stream>>>(
        sBuf, vBuf, edge_vec, senders, receivers, agsB, agvB, m0[l], m1[l], eTiles);
    node_update_kernel<<<nBlocks, WPB * 32, 0, stream>>>(
        sBuf, vBuf, agsB, agvB, nattr, ug[l], ul[l], nTiles);
  }

  decode_kernel<<<nBlocks, WPB * 32, 0, stream>>>(
      sBuf, vBuf, nattr, dec_gate, P(1, 2), P(1, 3), P(1, 4), (float*)d_out, nTiles);
}